// Behaviour_10247791968965
// MI455X (gfx1250) — compile-verified
//
#include <hip/hip_runtime.h>
#include <hip/hip_bf16.h>
#include <math.h>

typedef float v2f __attribute__((ext_vector_type(2)));
typedef float v8f __attribute__((ext_vector_type(8)));

__device__ __forceinline__ float sigf(float x) { return 1.0f / (1.0f + expf(-x)); }

__device__ __forceinline__ v8f wmma4(v2f a, v2f b, v8f c) {
  // D(16x16,f32) = A(16x4,f32) * B(4x16,f32) + C
  return __builtin_amdgcn_wmma_f32_16x16x4_f32(false, a, false, b, (short)0, c,
                                               false, false);
}

// ---------------------------------------------------------------------------
// K1: m1 = relu(x @ p1_W + p1_b); agg1 = 0        (N threads)
// ---------------------------------------------------------------------------
__global__ void __launch_bounds__(256) k_pool1(
    const float* __restrict__ x, const float* __restrict__ Wp,
    const float* __restrict__ bp, float* __restrict__ m, float* __restrict__ agg,
    int N) {
  int i = blockIdx.x * blockDim.x + threadIdx.x;
  if (i >= N) return;
  float v[5];
#pragma unroll
  for (int f = 0; f < 5; f++) v[f] = x[i * 5 + f];
#pragma unroll
  for (int j = 0; j < 5; j++) {
    float acc = bp[j];
#pragma unroll
    for (int f = 0; f < 5; f++) acc += v[f] * Wp[f * 5 + j];
    m[i * 5 + j] = fmaxf(acc, 0.0f);
    agg[i * 5 + j] = 0.0f;
  }
}

// ---------------------------------------------------------------------------
// K2/K4: agg[dst] = max(agg[dst], m[src]) per feature. m >= 0 (ReLU), so the
// f32 max equals an unsigned-int max on the bit pattern; init 0 gives the
// reference's empty-segment -> 0 semantics exactly. Emits non-returning
// GLOBAL_ATOMIC_MAX_U32 into the (L2-resident) agg array.
// ---------------------------------------------------------------------------
__global__ void __launch_bounds__(256) k_edge_max(
    const int* __restrict__ src, const int* __restrict__ dst,
    const float* __restrict__ m, float* __restrict__ agg, int E) {
  int e = blockIdx.x * blockDim.x + threadIdx.x;
  if (e >= E) return;
  int s = src[e] * 5;
  int d = dst[e] * 5;
  const unsigned* mp = (const unsigned*)m;
  unsigned* ap = (unsigned*)agg;
#pragma unroll
  for (int f = 0; f < 5; f++) atomicMax(ap + d + f, mp[s + f]);
}

// ---------------------------------------------------------------------------
// K3: h1 = sigmoid(x@s1W + agg1@n1W + b1); m2 = relu(h1@p2W + p2b); agg2 = 0
// (agg2 may alias agg1: per-thread loads precede the zero stores)
// ---------------------------------------------------------------------------
__global__ void __launch_bounds__(256) k_update1(
    const float* __restrict__ x, const float* agg1, const float* __restrict__ s1W,
    const float* __restrict__ n1W, const float* __restrict__ b1,
    const float* __restrict__ p2W, const float* __restrict__ p2b,
    float* __restrict__ h1out, float* __restrict__ m2, float* agg2, int N) {
  int i = blockIdx.x * blockDim.x + threadIdx.x;
  if (i >= N) return;
  float xv[5], av[5], h[5];
#pragma unroll
  for (int f = 0; f < 5; f++) { xv[f] = x[i * 5 + f]; av[f] = agg1[i * 5 + f]; }
#pragma unroll
  for (int j = 0; j < 5; j++) {
    float acc = b1[j];
#pragma unroll
    for (int f = 0; f < 5; f++) acc += xv[f] * s1W[f * 5 + j] + av[f] * n1W[f * 5 + j];
    h[j] = sigf(acc);
    h1out[i * 5 + j] = h[j];
  }
#pragma unroll
  for (int j = 0; j < 5; j++) {
    float acc = p2b[j];
#pragma unroll
    for (int f = 0; f < 5; f++) acc += h[f] * p2W[f * 5 + j];
    m2[i * 5 + j] = fmaxf(acc, 0.0f);
    agg2[i * 5 + j] = 0.0f;  // after all agg1 reads above
  }
}

// ---------------------------------------------------------------------------
// K5: per-leaf h2 = h1@s2W + agg2@n2W + b2; prod = h2*enc; then 4-layer MLP
// using V_WMMA_F32_16X16X4_F32. One wave = 32 leaves (two 16-row M tiles).
// ---------------------------------------------------------------------------
#define WPB 4  // waves per block (128 threads)

// Z = sigmoid(A[32x32] @ W[32x32] + b), A/Z in per-wave LDS (stride 34)
__device__ __forceinline__ void layer32(float (*A)[34], float (*Z)[34],
                                        const float* __restrict__ W,
                                        const float* __restrict__ bv, int lane) {
  int mrow = lane & 15;
  int kh = (lane >> 4) << 1;  // 0 or 2
  int col = lane & 15;
  float bias0 = bv[col], bias1 = bv[16 + col];
#pragma unroll
  for (int mt = 0; mt < 2; mt++) {
    v8f acc0, acc1;
#pragma unroll
    for (int r = 0; r < 8; r++) { acc0[r] = bias0; acc1[r] = bias1; }
#pragma unroll
    for (int kc = 0; kc < 8; kc++) {
      int kb = (kc << 2) + kh;
      v2f a;  a.x  = A[mt * 16 + mrow][kb];      a.y  = A[mt * 16 + mrow][kb + 1];
      v2f b0; b0.x = W[kb * 32 + col];           b0.y = W[(kb + 1) * 32 + col];
      v2f b1; b1.x = W[kb * 32 + 16 + col];      b1.y = W[(kb + 1) * 32 + 16 + col];
      acc0 = wmma4(a, b0, acc0);
      acc1 = wmma4(a, b1, acc1);
    }
    int rb = mt * 16 + ((lane >> 4) << 3);
#pragma unroll
    for (int r = 0; r < 8; r++) {
      Z[rb + r][col]      = sigf(acc0[r]);
      Z[rb + r][16 + col] = sigf(acc1[r]);
    }
  }
}

__global__ void __launch_bounds__(32 * WPB) k_final(
    const int* __restrict__ leaf, const float* __restrict__ h1,
    const float* __restrict__ agg2, const float* __restrict__ s2W,
    const float* __restrict__ n2W, const float* __restrict__ b2,
    const float* __restrict__ command, const float* __restrict__ cmdW,
    const float* __restrict__ cmdb, const float* __restrict__ o1W,
    const float* __restrict__ o1b, const float* __restrict__ o2W,
    const float* __restrict__ o2b, const float* __restrict__ o3W,
    const float* __restrict__ o3b, const float* __restrict__ o4W,
    const float* __restrict__ o4b, float* __restrict__ out, int L) {
  __shared__ float lds_x[WPB][32][12];
  __shared__ float lds_za[WPB][32][34];
  __shared__ float lds_zb[WPB][32][34];

  int lane = threadIdx.x & 31;
  int wid = threadIdx.x >> 5;
  int li = (blockIdx.x * WPB + wid) * 32 + lane;

  // ---- stage prod = (h1@s2W + agg2@n2W + b2) * enc into LDS [32 x 12] ----
  if (li < L) {
    float c0 = command[0], c1 = command[1];
    int node = leaf[li];
    float hv[5], av[5];
#pragma unroll
    for (int f = 0; f < 5; f++) {
      hv[f] = h1[node * 5 + f];
      av[f] = agg2[node * 5 + f];
    }
#pragma unroll
    for (int j = 0; j < 10; j++) {
      float h2 = b2[j];
#pragma unroll
      for (int f = 0; f < 5; f++) h2 += hv[f] * s2W[f * 10 + j] + av[f] * n2W[f * 10 + j];
      float enc = c0 * cmdW[j] + c1 * cmdW[10 + j] + cmdb[j];
      lds_x[wid][lane][j] = h2 * enc;
    }
    lds_x[wid][lane][10] = 0.0f;
    lds_x[wid][lane][11] = 0.0f;
  } else {
#pragma unroll
    for (int j = 0; j < 12; j++) lds_x[wid][lane][j] = 0.0f;
  }
  __syncthreads();

  // ---- layer 1: [32x10] @ o1W[10x32] (K padded to 12) -> za ----
  {
    int mrow = lane & 15;
    int kh = (lane >> 4) << 1;
    int col = lane & 15;
    float bias0 = o1b[col], bias1 = o1b[16 + col];
#pragma unroll
    for (int mt = 0; mt < 2; mt++) {
      v8f acc0, acc1;
#pragma unroll
      for (int r = 0; r < 8; r++) { acc0[r] = bias0; acc1[r] = bias1; }
#pragma unroll
      for (int kc = 0; kc < 3; kc++) {
        int kb = (kc << 2) + kh;
        v2f a;  a.x  = lds_x[wid][mt * 16 + mrow][kb];
                a.y  = lds_x[wid][mt * 16 + mrow][kb + 1];
        v2f b0; b0.x = (kb < 10)     ? o1W[kb * 32 + col] : 0.0f;
                b0.y = (kb + 1 < 10) ? o1W[(kb + 1) * 32 + col] : 0.0f;
        v2f b1; b1.x = (kb < 10)     ? o1W[kb * 32 + 16 + col] : 0.0f;
                b1.y = (kb + 1 < 10) ? o1W[(kb + 1) * 32 + 16 + col] : 0.0f;
        acc0 = wmma4(a, b0, acc0);
        acc1 = wmma4(a, b1, acc1);
      }
      int rb = mt * 16 + ((lane >> 4) << 3);
#pragma unroll
      for (int r = 0; r < 8; r++) {
        lds_za[wid][rb + r][col]      = sigf(acc0[r]);
        lds_za[wid][rb + r][16 + col] = sigf(acc1[r]);
      }
    }
  }
  __syncthreads();

  layer32(lds_za[wid], lds_zb[wid], o2W, o2b, lane);  // layer 2
  __syncthreads();
  layer32(lds_zb[wid], lds_za[wid], o3W, o3b, lane);  // layer 3
  __syncthreads();

  // ---- layer 4: 32 -> 1 dot per leaf ----
  if (li < L) {
    float acc = o4b[0];
#pragma unroll
    for (int k = 0; k < 32; k++) acc += lds_za[wid][lane][k] * o4W[k];
    out[li] = sigf(acc);
  }
}

// ---------------------------------------------------------------------------
extern "C" void kernel_launch(void* const* d_in, const int* in_sizes, int n_in,
                              void* d_out, int out_size, void* d_ws, size_t ws_size,
                              hipStream_t stream) {
  const float* x       = (const float*)d_in[0];
  const int*   src     = (const int*)d_in[1];
  const int*   dst     = (const int*)d_in[2];
  const int*   leaf    = (const int*)d_in[3];
  const float* command = (const float*)d_in[4];
  const float* p1W = (const float*)d_in[5];
  const float* p1b = (const float*)d_in[6];
  const float* s1W = (const float*)d_in[7];
  const float* n1W = (const float*)d_in[8];
  const float* b1  = (const float*)d_in[9];
  const float* p2W = (const float*)d_in[10];
  const float* p2b = (const float*)d_in[11];
  const float* s2W = (const float*)d_in[12];
  const float* n2W = (const float*)d_in[13];
  const float* b2  = (const float*)d_in[14];
  const float* cmdW = (const float*)d_in[15];
  const float* cmdb = (const float*)d_in[16];
  const float* o1W = (const float*)d_in[17];
  const float* o1b = (const float*)d_in[18];
  const float* o2W = (const float*)d_in[19];
  const float* o2b = (const float*)d_in[20];
  const float* o3W = (const float*)d_in[21];
  const float* o3b = (const float*)d_in[22];
  const float* o4W = (const float*)d_in[23];
  const float* o4b = (const float*)d_in[24];

  int N = in_sizes[0] / 5;
  int E = in_sizes[1];
  int L = in_sizes[3];

  float* ws  = (float*)d_ws;
  float* m   = ws;                      // N*5 : m1 then m2 (m1 dead after edge pass 1)
  float* agg = ws + (size_t)N * 5;      // N*5 : agg1 then agg2 (safe per-thread alias)
  float* h1  = ws + (size_t)N * 10;     // N*5

  int tb = 256;
  k_pool1<<<(N + tb - 1) / tb, tb, 0, stream>>>(x, p1W, p1b, m, agg, N);
  k_edge_max<<<(E + tb - 1) / tb, tb, 0, stream>>>(src, dst, m, agg, E);
  k_update1<<<(N + tb - 1) / tb, tb, 0, stream>>>(x, agg, s1W, n1W, b1, p2W, p2b,
                                                  h1, m, agg, N);
  k_edge_max<<<(E + tb - 1) / tb, tb, 0, stream>>>(src, dst, m, agg, E);

  int lpb = 32 * WPB;  // leaves per block
  k_final<<<(L + lpb - 1) / lpb, 32 * WPB, 0, stream>>>(
      leaf, h1, agg, s2W, n2W, b2, command, cmdW, cmdb, o1W, o1b, o2W, o2b, o3W,
      o3b, o4W, o4b, (float*)d_out, L);
}